// CRD_CLS_16028817949561
// MI455X (gfx1250) — compile-verified
//
#include <hip/hip_runtime.h>
#include <hip/hip_bf16.h>
#include <math.h>

// ---------------------------------------------------------------------------
// 2-layer GCN (DGL GraphConv norm='both') for MI455X / gfx1250, wave32.
// GEMMs use V_WMMA_F32_16X16X4_F32 (full f32 precision matrix core path).
// ---------------------------------------------------------------------------

#define GCN_N     100000
#define GCN_E     1600000
#define GCN_FIN   128
#define GCN_HID   64
#define GCN_C     40
#define KEEP_P    0.8f

typedef __attribute__((ext_vector_type(2))) float v2f;
typedef __attribute__((ext_vector_type(8))) float v8f;

// ----------------------------- utility kernels -----------------------------

__global__ void gcn_zero_f32(float* __restrict__ p, int n) {
  int stride = gridDim.x * blockDim.x;
  for (int i = blockIdx.x * blockDim.x + threadIdx.x; i < n; i += stride)
    p[i] = 0.0f;
}

__global__ void gcn_degrees(const int* __restrict__ src, const int* __restrict__ dst,
                            float* __restrict__ deg_out, float* __restrict__ deg_in, int e) {
  int i = blockIdx.x * blockDim.x + threadIdx.x;
  if (i >= e) return;
  atomicAdd(&deg_out[src[i]], 1.0f);
  atomicAdd(&deg_in[dst[i]], 1.0f);
}

__global__ void gcn_norms(const float* __restrict__ deg_out, const float* __restrict__ deg_in,
                          float* __restrict__ norm_out, float* __restrict__ norm_in, int n) {
  int i = blockIdx.x * blockDim.x + threadIdx.x;
  if (i >= n) return;
  norm_out[i] = rsqrtf(fmaxf(deg_out[i], 1.0f));
  norm_in[i]  = rsqrtf(fmaxf(deg_in[i],  1.0f));
}

// ----------------------- WMMA GEMM, layer 1 (K=128) ------------------------
// Y[N,64] = (X[N,128] * norm_out[:,None]) @ W[128,64]
// One wave per 16x16 output tile. f32 WMMA 16x16x4, 32 K-steps.
// A-frag (16x4 f32): lanes 0-15 carry M=lane, K-pair {0,1}; lanes 16-31
// carry M=lane-16, K-pair {2,3}. B-frag mirrors K split; lane gives N.
__global__ void gcn_gemm1_wmma(const float* __restrict__ X,
                               const float* __restrict__ norm_out,
                               const float* __restrict__ W,
                               float* __restrict__ Y) {
  const int ntiles_n = GCN_HID / 16;                 // 4
  const int tiles    = (GCN_N / 16) * ntiles_n;      // 25000
  int wave = blockIdx.x * (blockDim.x >> 5) + (threadIdx.x >> 5);
  if (wave >= tiles) return;                         // wave-uniform
  int tm = wave / ntiles_n;
  int tn = wave % ntiles_n;

  int lane = threadIdx.x & 31;
  int half = lane >> 4;          // 0: K offsets {0,1}; 1: K offsets {2,3}
  int l    = lane & 15;
  int row  = tm * 16 + l;        // A row for this lane
  int col  = tn * 16 + l;        // B/D column for this lane
  float nrm = norm_out[row];

  v8f acc = {};
  const float* xr = X + (long long)row * GCN_FIN;
#pragma unroll 4
  for (int k4 = 0; k4 < GCN_FIN / 4; ++k4) {
    int kb = k4 * 4 + half * 2;
    v2f a, b;
    a.x = xr[kb + 0] * nrm;
    a.y = xr[kb + 1] * nrm;
    b.x = W[(kb + 0) * GCN_HID + col];
    b.y = W[(kb + 1) * GCN_HID + col];
    acc = __builtin_amdgcn_wmma_f32_16x16x4_f32(false, a, false, b,
                                                (short)0, acc, false, false);
  }
  // D layout: VGPR j -> M = j (lanes 0-15) / M = 8+j (lanes 16-31), N = l
  int rbase = tm * 16 + half * 8;
#pragma unroll
  for (int j = 0; j < 8; ++j)
    Y[(long long)(rbase + j) * GCN_HID + col] = acc[j];
}

// --------------------------- edge scatter, layer 1 -------------------------
__global__ void gcn_scatter64(const int* __restrict__ src, const int* __restrict__ dst,
                              const float* __restrict__ Y, float* __restrict__ H) {
  int tid = blockIdx.x * blockDim.x + threadIdx.x;   // E*64 threads
  if (tid >= GCN_E * GCN_HID) return;
  int e = tid >> 6;
  int f = tid & 63;
  int s = src[e], d = dst[e];
  atomicAdd(&H[(long long)d * GCN_HID + f], Y[(long long)s * GCN_HID + f]);
}

// ------------------- bias + norm_in + relu + dropout -----------------------
__device__ __forceinline__ unsigned gcn_hash(unsigned x) {
  x ^= x >> 16; x *= 0x7feb352dU;
  x ^= x >> 15; x *= 0x846ca68bU;
  x ^= x >> 16; return x;
}

__global__ void gcn_post1(const float* __restrict__ H, const float* __restrict__ norm_in,
                          const float* __restrict__ b1, float* __restrict__ X2) {
  int tid = blockIdx.x * blockDim.x + threadIdx.x;   // N*64 threads
  if (tid >= GCN_N * GCN_HID) return;
  int node = tid >> 6;
  int f    = tid & 63;
  float v = H[tid] * norm_in[node] + b1[f];
  v = fmaxf(v, 0.0f);                                // relu
  // deterministic dropout mask (keep prob 0.8), scale by 1/0.8
  unsigned h = gcn_hash((unsigned)tid ^ 0x9E3779B9u);
  bool keep = (h >> 8) < (unsigned)(KEEP_P * 16777216.0f);
  X2[tid] = keep ? v * (1.0f / KEEP_P) : 0.0f;
}

// ----------------------- WMMA GEMM, layer 2 (K=64) -------------------------
// Y2[N,40] = X2[N,64] @ W2[64,40]; N-tiles of 16 (3 tiles, last half-masked)
__global__ void gcn_gemm2_wmma(const float* __restrict__ X2,
                               const float* __restrict__ W2,
                               float* __restrict__ Y2) {
  const int ntiles_n = 3;                            // ceil(40/16)
  const int tiles    = (GCN_N / 16) * ntiles_n;      // 18750
  int wave = blockIdx.x * (blockDim.x >> 5) + (threadIdx.x >> 5);
  if (wave >= tiles) return;                         // wave-uniform
  int tm = wave / ntiles_n;
  int tn = wave % ntiles_n;

  int lane = threadIdx.x & 31;
  int half = lane >> 4;
  int l    = lane & 15;
  int row  = tm * 16 + l;
  int col  = tn * 16 + l;
  bool colok = (col < GCN_C);

  v8f acc = {};
  const float* xr = X2 + (long long)row * GCN_HID;
#pragma unroll 4
  for (int k4 = 0; k4 < GCN_HID / 4; ++k4) {
    int kb = k4 * 4 + half * 2;
    v2f a, b;
    a.x = xr[kb + 0];
    a.y = xr[kb + 1];
    b.x = colok ? W2[(kb + 0) * GCN_C + col] : 0.0f;
    b.y = colok ? W2[(kb + 1) * GCN_C + col] : 0.0f;
    acc = __builtin_amdgcn_wmma_f32_16x16x4_f32(false, a, false, b,
                                                (short)0, acc, false, false);
  }
  if (colok) {
    int rbase = tm * 16 + half * 8;
#pragma unroll
    for (int j = 0; j < 8; ++j)
      Y2[(long long)(rbase + j) * GCN_C + col] = acc[j];
  }
}

// --------------------------- edge scatter, layer 2 -------------------------
__global__ void gcn_scatter40(const int* __restrict__ src, const int* __restrict__ dst,
                              const float* __restrict__ Y2, float* __restrict__ out) {
  int tid = blockIdx.x * blockDim.x + threadIdx.x;   // E*40 threads
  if (tid >= GCN_E * GCN_C) return;
  int e = tid / GCN_C;
  int f = tid - e * GCN_C;
  int s = src[e], d = dst[e];
  atomicAdd(&out[(long long)d * GCN_C + f], Y2[(long long)s * GCN_C + f]);
}

// ------------------- norm_in + bias + log_softmax (in place) ---------------
__global__ void gcn_logsoftmax(float* __restrict__ out, const float* __restrict__ norm_in,
                               const float* __restrict__ b2) {
  int i = blockIdx.x * blockDim.x + threadIdx.x;
  if (i >= GCN_N) return;
  float ni = norm_in[i];
  float* row = out + (long long)i * GCN_C;
  float v[GCN_C];
  float m = -INFINITY;
#pragma unroll
  for (int c = 0; c < GCN_C; ++c) {
    v[c] = row[c] * ni + b2[c];
    m = fmaxf(m, v[c]);
  }
  float s = 0.0f;
#pragma unroll
  for (int c = 0; c < GCN_C; ++c) s += __expf(v[c] - m);
  float lg = __logf(s);
#pragma unroll
  for (int c = 0; c < GCN_C; ++c) row[c] = v[c] - m - lg;
}

// --------------------------------- launcher --------------------------------

extern "C" void kernel_launch(void* const* d_in, const int* in_sizes, int n_in,
                              void* d_out, int out_size, void* d_ws, size_t ws_size,
                              hipStream_t stream) {
  const float* in_feat = (const float*)d_in[0];         // [N,128]
  const int*   edges   = (const int*)d_in[1];           // [2,E]
  const float* W1      = (const float*)d_in[2];         // [128,64]
  const float* b1      = (const float*)d_in[3];         // [64]
  const float* W2      = (const float*)d_in[4];         // [64,40]
  const float* b2      = (const float*)d_in[5];         // [40]
  float*       out     = (float*)d_out;                 // [N,40]
  (void)in_sizes; (void)n_in; (void)out_size; (void)ws_size;

  const int* src = edges;
  const int* dst = edges + GCN_E;

  // workspace layout (floats): deg_out | deg_in | norm_out | norm_in | y1/x2 | h1/y2
  float* ws       = (float*)d_ws;
  float* deg_out  = ws;
  float* deg_in   = ws + (size_t)GCN_N;
  float* norm_out = ws + (size_t)2 * GCN_N;
  float* norm_in  = ws + (size_t)3 * GCN_N;
  float* y1       = ws + (size_t)4 * GCN_N;                       // N*64 (reused as x2)
  float* h1       = ws + (size_t)4 * GCN_N + (size_t)GCN_N * 64;  // N*64 (reused as y2)

  const int B = 256;

  // 1) zero accumulators (d_out + d_ws are poisoned by the harness)
  gcn_zero_f32<<<2048, B, 0, stream>>>(deg_out, 2 * GCN_N);
  gcn_zero_f32<<<4096, B, 0, stream>>>(h1, GCN_N * GCN_HID);
  gcn_zero_f32<<<4096, B, 0, stream>>>(out, GCN_N * GCN_C);

  // 2) degrees, 3) norms
  gcn_degrees<<<(GCN_E + B - 1) / B, B, 0, stream>>>(src, dst, deg_out, deg_in, GCN_E);
  gcn_norms<<<(GCN_N + B - 1) / B, B, 0, stream>>>(deg_out, deg_in, norm_out, norm_in, GCN_N);

  // 4) layer-1 GEMM (WMMA f32), norm_out fused
  {
    int tiles = (GCN_N / 16) * (GCN_HID / 16);   // 25000 waves
    int waves_per_blk = 128 / 32;
    int blocks = (tiles + waves_per_blk - 1) / waves_per_blk;
    gcn_gemm1_wmma<<<blocks, 128, 0, stream>>>(in_feat, norm_out, W1, y1);
  }

  // 5) scatter-sum layer 1
  gcn_scatter64<<<(GCN_E * GCN_HID + B - 1) / B, B, 0, stream>>>(src, dst, y1, h1);

  // 6) norm_in + bias + relu + dropout  (writes x2 over y1 buffer)
  gcn_post1<<<(GCN_N * GCN_HID + B - 1) / B, B, 0, stream>>>(h1, norm_in, b1, y1);

  // 7) layer-2 GEMM (WMMA f32)  (writes y2 over h1 buffer)
  {
    int tiles = (GCN_N / 16) * 3;                // 18750 waves
    int waves_per_blk = 128 / 32;
    int blocks = (tiles + waves_per_blk - 1) / waves_per_blk;
    gcn_gemm2_wmma<<<blocks, 128, 0, stream>>>(y1, W2, h1);
  }

  // 8) scatter-sum layer 2 directly into d_out
  gcn_scatter40<<<(GCN_E * GCN_C + B - 1) / B, B, 0, stream>>>(src, dst, h1, out);

  // 9) norm_in + bias + log_softmax in place
  gcn_logsoftmax<<<(GCN_N + B - 1) / B, B, 0, stream>>>(out, norm_in, b2);
}